// GNN_80607946211347
// MI455X (gfx1250) — compile-verified
//
#include <hip/hip_runtime.h>

#define IN_DIM 128
#define HID_DIM 128

typedef float v2f __attribute__((ext_vector_type(2)));
typedef float v8f __attribute__((ext_vector_type(8)));

// ---------------------------------------------------------------------------
// Init: deg = 1 (self loop), agg = 0, out = 0
// ---------------------------------------------------------------------------
__global__ void gcn_init(float* __restrict__ deg, float* __restrict__ agg,
                         float* __restrict__ out, int N) {
    long long total = (long long)N * HID_DIM;
    for (long long i = (long long)blockIdx.x * blockDim.x + threadIdx.x;
         i < total; i += (long long)gridDim.x * blockDim.x) {
        agg[i] = 0.0f;
        if (i < N) { deg[i] = 1.0f; out[i] = 0.0f; }
    }
}

// ---------------------------------------------------------------------------
// Degree accumulation over edge destinations
// ---------------------------------------------------------------------------
__global__ void gcn_degree(const int* __restrict__ ei, float* __restrict__ deg,
                           int E) {
    int e = blockIdx.x * blockDim.x + threadIdx.x;
    if (e < E) atomicAdd(deg + ei[E + e], 1.0f);
}

// deg -> deg^{-1/2} in place
__global__ void gcn_rsqrt(float* __restrict__ dinv, int N) {
    int i = blockIdx.x * blockDim.x + threadIdx.x;
    if (i < N) {
        float d = dinv[i];
        dinv[i] = (d > 0.0f) ? rsqrtf(d) : 0.0f;
    }
}

// ---------------------------------------------------------------------------
// H = X @ W  (f32 WMMA 16x16x4). One wave per 16-row x 128-col stripe:
// 8 independent 16x16 accumulators, A fragment loaded once per k-step and
// reused across all 8 column tiles (8 back-to-back WMMAs per A load).
//
// Fragment maps (ISA 7.12.2, 32-bit):
//   lane = 16*half + r
//   A (16x4):  a = { X[m0+r][k+2h], X[m0+r][k+2h+1] }      (one b64 load)
//   B (4x16):  b = { W[k+2h][n0+r], W[k+2h+1][n0+r] }
//   C/D:       VGPR j -> row m0 + 8*half + j, col n0 + r
// A row m only contributes to D row m, so OOB A rows are clamped (their D
// rows are never stored) -- keeps EXEC all-ones with no per-iter branching.
// ---------------------------------------------------------------------------
__global__ void gcn_gemm_wmma(const float* __restrict__ X,
                              const float* __restrict__ W,
                              float* __restrict__ H, int N) {
    int wave = (int)((blockIdx.x * (long long)blockDim.x + threadIdx.x) >> 5);
    int lane = threadIdx.x & 31;
    int numRowTiles = (N + 15) >> 4;
    if (wave >= numRowTiles) return;         // wave-uniform guard: EXEC stays full
    int m0 = wave * 16;
    int half = lane >> 4, r = lane & 15;

    int arow = m0 + r;
    if (arow >= N) arow = N - 1;             // safe clamp (see note above)
    const float* aptr = X + (size_t)arow * IN_DIM + 2 * half;
    const float* bbase = W + (size_t)(2 * half) * HID_DIM + r;

    v8f acc[8];
#pragma unroll
    for (int ct = 0; ct < 8; ++ct) acc[ct] = v8f{};

#pragma unroll 4
    for (int k = 0; k < IN_DIM; k += 4) {
        v2f a = *(const v2f*)(aptr + k);
        const float* bk = bbase + (size_t)k * HID_DIM;
#pragma unroll
        for (int ct = 0; ct < 8; ++ct) {
            v2f b;
            b.x = bk[16 * ct];
            b.y = bk[HID_DIM + 16 * ct];
            acc[ct] = __builtin_amdgcn_wmma_f32_16x16x4_f32(
                          /*neg_a=*/false, a, /*neg_b=*/false, b,
                          /*c_mod=*/(short)0, acc[ct],
                          /*reuse_a=*/false, /*reuse_b=*/false);
        }
    }

    int orow0 = m0 + 8 * half;
#pragma unroll
    for (int ct = 0; ct < 8; ++ct) {
#pragma unroll
        for (int j = 0; j < 8; ++j) {
            int row = orow0 + j;
            if (row < N) H[(size_t)row * HID_DIM + 16 * ct + r] = acc[ct][j];
        }
    }
}

// ---------------------------------------------------------------------------
// Layer-1 edge scatter: one wave per edge, float4 per lane (32*4 = 128 feats)
// agg[dst] += h[src] * dinv[src]*dinv[dst]
// ---------------------------------------------------------------------------
__global__ void gcn_edge_scatter128(const int* __restrict__ ei,
                                    const float* __restrict__ h,
                                    const float* __restrict__ dinv,
                                    float* __restrict__ agg, int E) {
    int lane = threadIdx.x & 31;
    int e = (int)((blockIdx.x * (long long)blockDim.x + threadIdx.x) >> 5);
    if (e >= E) return;
    int s = ei[e];
    int d = ei[E + e];
    float nrm = dinv[s] * dinv[d];
    float4 v = ((const float4*)(h + (size_t)s * HID_DIM))[lane];
    float* ad = agg + (size_t)d * HID_DIM + lane * 4;
    atomicAdd(ad + 0, v.x * nrm);
    atomicAdd(ad + 1, v.y * nrm);
    atomicAdd(ad + 2, v.z * nrm);
    atomicAdd(ad + 3, v.w * nrm);
}

// ---------------------------------------------------------------------------
// Self-loop + bias + ReLU (in place on agg):
// agg[n,f] = relu(agg[n,f] + h[n,f]*dinv[n]^2 + b1[f])
// ---------------------------------------------------------------------------
__global__ void gcn_self_bias_relu(const float* __restrict__ h,
                                   const float* __restrict__ dinv,
                                   const float* __restrict__ b1,
                                   float* __restrict__ agg, int N) {
    long long total = (long long)N * HID_DIM;
    long long i = (long long)blockIdx.x * blockDim.x + threadIdx.x;
    if (i >= total) return;
    int n = (int)(i >> 7);        // /128
    int f = (int)(i & 127);
    float di = dinv[n];
    float v = agg[i] + h[i] * di * di + b1[f];
    agg[i] = v > 0.0f ? v : 0.0f;
}

// ---------------------------------------------------------------------------
// t2[n] = dot(agg[n,:], W2[:,0]); one wave per node, shfl_xor reduction
// ---------------------------------------------------------------------------
__global__ void gcn_dot_w2(const float* __restrict__ agg,
                           const float* __restrict__ W2,
                           float* __restrict__ t2, int N) {
    int lane = threadIdx.x & 31;
    int n = (int)((blockIdx.x * (long long)blockDim.x + threadIdx.x) >> 5);
    if (n >= N) return;
    float4 v = ((const float4*)(agg + (size_t)n * HID_DIM))[lane];
    float4 w = ((const float4*)W2)[lane];
    float s = v.x * w.x + v.y * w.y + v.z * w.z + v.w * w.w;
#pragma unroll
    for (int off = 16; off > 0; off >>= 1) s += __shfl_xor(s, off, 32);
    if (lane == 0) t2[n] = s;
}

// ---------------------------------------------------------------------------
// Layer-2 edge scatter (scalar): out[dst] += t2[src] * dinv[src]*dinv[dst]
// ---------------------------------------------------------------------------
__global__ void gcn_edge_scatter1(const int* __restrict__ ei,
                                  const float* __restrict__ t2,
                                  const float* __restrict__ dinv,
                                  float* __restrict__ out, int E) {
    int e = blockIdx.x * blockDim.x + threadIdx.x;
    if (e >= E) return;
    int s = ei[e];
    int d = ei[E + e];
    atomicAdd(out + d, t2[s] * dinv[s] * dinv[d]);
}

// out[i] += t2[i]*dinv[i]^2 + b2
__global__ void gcn_finalize(const float* __restrict__ t2,
                             const float* __restrict__ dinv,
                             const float* __restrict__ b2,
                             float* __restrict__ out, int N) {
    int i = blockIdx.x * blockDim.x + threadIdx.x;
    if (i >= N) return;
    float di = dinv[i];
    out[i] += t2[i] * di * di + b2[0];
}

// ---------------------------------------------------------------------------
extern "C" void kernel_launch(void* const* d_in, const int* in_sizes, int n_in,
                              void* d_out, int out_size, void* d_ws, size_t ws_size,
                              hipStream_t stream) {
    const float* x   = (const float*)d_in[0];
    const int*   ei  = (const int*)d_in[1];
    const float* W1  = (const float*)d_in[2];
    const float* b1  = (const float*)d_in[3];
    const float* W2  = (const float*)d_in[4];
    const float* b2  = (const float*)d_in[5];
    float* out = (float*)d_out;

    const int N = in_sizes[0] / IN_DIM;
    const int E = in_sizes[1] / 2;

    // Workspace layout (floats): dinv[N] | h[N*128] | agg[N*128] | t2[N]
    float* dinv = (float*)d_ws;
    float* h    = dinv + N;
    float* agg  = h + (size_t)N * HID_DIM;
    float* t2   = agg + (size_t)N * HID_DIM;

    const int B = 256;
    long long totalNF = (long long)N * HID_DIM;
    int blocksNF = (int)((totalNF + B - 1) / B);

    // 1. init deg=1, agg=0, out=0
    gcn_init<<<blocksNF, B, 0, stream>>>(dinv, agg, out, N);
    // 2. degree
    gcn_degree<<<(E + B - 1) / B, B, 0, stream>>>(ei, dinv, E);
    // 3. deg^{-1/2}
    gcn_rsqrt<<<(N + B - 1) / B, B, 0, stream>>>(dinv, N);
    // 4. h = x @ W1  (WMMA f32, one wave per 16x128 stripe)
    {
        int numRowTiles = (N + 15) / 16;
        int wavesPerBlock = B / 32;
        int blocks = (numRowTiles + wavesPerBlock - 1) / wavesPerBlock;
        gcn_gemm_wmma<<<blocks, B, 0, stream>>>(x, W1, h, N);
    }
    // 5. edge scatter layer 1 (wave per edge)
    {
        int wavesPerBlock = B / 32;
        int blocks = (E + wavesPerBlock - 1) / wavesPerBlock;
        gcn_edge_scatter128<<<blocks, B, 0, stream>>>(ei, h, dinv, agg, E);
    }
    // 6. self-loop + bias + relu
    gcn_self_bias_relu<<<blocksNF, B, 0, stream>>>(h, dinv, b1, agg, N);
    // 7. t2 = agg @ W2 (wave per node)
    {
        int wavesPerBlock = B / 32;
        int blocks = (N + wavesPerBlock - 1) / wavesPerBlock;
        gcn_dot_w2<<<blocks, B, 0, stream>>>(agg, W2, t2, N);
    }
    // 8. edge scatter layer 2 (scalar)
    gcn_edge_scatter1<<<(E + B - 1) / B, B, 0, stream>>>(ei, t2, dinv, out, E);
    // 9. self-loop + bias
    gcn_finalize<<<(N + B - 1) / B, B, 0, stream>>>(t2, dinv, b2, out, N);
}